// DoubleConvolutionalHead_31980326486578
// MI455X (gfx1250) — compile-verified
//
#include <hip/hip_runtime.h>

typedef float v2f __attribute__((ext_vector_type(2)));
typedef float v8f __attribute__((ext_vector_type(8)));

static constexpr int  kN   = 8;
static constexpr int  kTP  = 2048;
static constexpr int  kE   = 4096;
static constexpr int  kT   = 16384;     // TP * S
static constexpr int  kV   = 17;        // V+1
static constexpr long long kOutPerN = 2228224LL; // T*S*V = 131072*17
static constexpr int  kBinCap = 2048;   // max entries per (n,gs)

// ---------------------------------------------------------------------------
// Kernel 1: per-sample truncation index + stable compaction rank + gs-binning
// value/depth are int64; we read the low 32-bit word (values are small, >=0).
// ---------------------------------------------------------------------------
__global__ void __launch_bounds__(256)
scan_kernel(const int* __restrict__ val32, const int* __restrict__ dep32,
            unsigned* __restrict__ bins, unsigned* __restrict__ bincnt) {
  __shared__ unsigned partial[256];
  __shared__ unsigned lbin[8];
  const int n   = blockIdx.x;
  const int tid = threadIdx.x;
  const int* dep = dep32 + (size_t)2 * n * kT;
  const int* val = val32 + (size_t)2 * n * kT;

  // depth is sorted ascending: argmax == lower_bound(max value)
  const int maxv = dep[2 * (kT - 1)];
  int lo = 0, hi = kT;
  while (lo < hi) {
    int mid = (lo + hi) >> 1;
    if (dep[2 * mid] < maxv) lo = mid + 1; else hi = mid;
  }
  const int idx = lo;

  // pass 1: count selected in my contiguous chunk of 64
  const int t0 = tid * 64;
  unsigned cnt = 0;
  for (int j = 0; j < 64; ++j) {
    int t = t0 + j;
    cnt += (t < idx && val[2 * t] == 2) ? 1u : 0u;
  }
  partial[tid] = cnt;
  __syncthreads();
  // Hillis-Steele inclusive scan over 256 partials
  for (int d = 1; d < 256; d <<= 1) {
    unsigned v = (tid >= d) ? partial[tid - d] : 0u;
    __syncthreads();
    partial[tid] += v;
    __syncthreads();
  }
  unsigned off = partial[tid] - cnt;  // exclusive
  if (tid < 8) lbin[tid] = 0;
  __syncthreads();

  // pass 2: assign ranks in order, bin by gs = t & 7, pack (gp<<14)|r
  for (int j = 0; j < 64; ++j) {
    int t = t0 + j;
    if (t < idx && val[2 * t] == 2) {
      unsigned r  = off++;
      unsigned gs = (unsigned)(t & 7);
      unsigned slot = atomicAdd(&lbin[gs], 1u);
      bins[((unsigned)(n * 8) + gs) * kBinCap + slot] =
          ((unsigned)(t >> 3) << 14) | r;
    }
  }
  __syncthreads();
  if (tid < 8) bincnt[n * 8 + tid] = lbin[tid];
}

// ---------------------------------------------------------------------------
// Kernel 2: Wfused[e,gs,s] = sum_c W1[e,c,gs]*W2[c,s];  bias[s]=b2+sum b1*W2
// ---------------------------------------------------------------------------
__global__ void __launch_bounds__(256)
fuse_kernel(const float* __restrict__ W1, const float* __restrict__ b1,
            const float* __restrict__ W2, const float* __restrict__ b2,
            float* __restrict__ wf, float* __restrict__ bias8) {
  const int o  = blockIdx.x * 256 + threadIdx.x;   // 0 .. 262143
  const int e  = o >> 6;
  const int gs = (o >> 3) & 7;
  const int s  = o & 7;
  float acc = 0.f;
  const float* w1e = W1 + (size_t)e * 512;
  for (int c = 0; c < 64; ++c) acc += w1e[c * 8 + gs] * W2[c * 8 + s];
  wf[o] = acc;
  if (o < 8) {
    float b = b2[0];
    for (int c = 0; c < 64; ++c) b += b1[c] * W2[c * 8 + o];
    bias8[o] = b;
  }
}

// ---------------------------------------------------------------------------
// Kernel 3: fill entire output with constant b2 * W3[v]
// ---------------------------------------------------------------------------
__global__ void __launch_bounds__(256)
fill_kernel(const float* __restrict__ b2, const float* __restrict__ W3,
            float* __restrict__ out) {
  long long i = (long long)blockIdx.x * 256 + threadIdx.x;
  if (i < (long long)kN * kOutPerN) out[i] = b2[0] * W3[(int)(i % 17)];
}

// ---------------------------------------------------------------------------
// Kernel 4: main WMMA kernel. Block = (n,gs); 8 waves, each does 16-row tiles.
//   z[16x8] = Xrows[16x4096] x Wfused[:,gs,:][4096x8]  via v_wmma_f32_16x16x4
//   out[n, r*8+s, v] = (z + bias[s]) * W3[v]
// ---------------------------------------------------------------------------
__global__ void __launch_bounds__(256)
main_kernel(const float* __restrict__ x, const float* __restrict__ wf,
            const float* __restrict__ bias8, const float* __restrict__ W3,
            const unsigned* __restrict__ bins,
            const unsigned* __restrict__ bincnt, float* __restrict__ out) {
  __shared__ float Bt[8 * 4096];   // B transposed: Bt[s][k], 128 KB
  __shared__ float Zpad[4104];     // zero source for WMMA cols 8..15
  __shared__ float zst[8][128];    // per-wave 16x8 result staging
  __shared__ float w3s[17];

  const int pair = blockIdx.x;          // 0..63
  const int n = pair >> 3, gs = pair & 7;
  const int tid = threadIdx.x;

  // stage Bt: Bt[s*4096+k] = Wfused[k*64 + gs*8 + s]
  for (int i = tid; i < 8 * 4096; i += 256) {
    int k = i >> 3, s = i & 7;
    Bt[s * 4096 + k] = wf[(size_t)k * 64 + gs * 8 + s];
  }
  for (int i = tid; i < 4104; i += 256) Zpad[i] = 0.f;
  if (tid < 17) w3s[tid] = W3[tid];
  __syncthreads();

  const unsigned cnt = bincnt[n * 8 + gs];
  const int nTiles = (int)((cnt + 15u) >> 4);
  const int wave = tid >> 5, lane = tid & 31;
  const int n16 = lane & 15, half = lane >> 4;   // half selects K pair / M block
  const unsigned* mybins = bins + (unsigned)(n * 8 + gs) * kBinCap;
  const float* xs = x + (size_t)n * kTP * kE;
  const float biasv = (n16 < 8) ? bias8[n16] : 0.f;
  const float* bbase = ((n16 < 8) ? (Bt + n16 * 4096) : Zpad) + half * 2;

  // tiles strided across 8 blockIdx.y slices * 8 waves
  for (int tile = blockIdx.y * 8 + wave; tile < nTiles; tile += 64) {
    const unsigned eidx = (unsigned)tile * 16u + (unsigned)n16;
    const unsigned ep = (eidx < cnt) ? mybins[eidx] : 0xFFFFFFFFu;
    const unsigned gp = (ep == 0xFFFFFFFFu) ? 0u : (ep >> 14);
    const float* arow = xs + (size_t)gp * kE + half * 2;

    v8f acc = {0.f, 0.f, 0.f, 0.f, 0.f, 0.f, 0.f, 0.f};
#pragma unroll 8
    for (int k = 0; k < kE; k += 4) {
      v2f a = *(const v2f*)(arow + k);    // A: lane(M=n16) K = k+2*half .. +1
      v2f b = *(const v2f*)(bbase + k);   // B: lane(N=n16) K = k+2*half .. +1
      acc = __builtin_amdgcn_wmma_f32_16x16x4_f32(
          false, a, false, b, (short)0, acc, false, false);
    }

    // D layout: lane, vgpr j -> M = j + 8*half, N = n16
    if (n16 < 8) {
#pragma unroll
      for (int j = 0; j < 8; ++j)
        zst[wave][(j + half * 8) * 8 + n16] = acc[j] + biasv;
    }

    // scatter: each valid rank owns 8*17 = 136 contiguous output floats
    for (int m = 0; m < 16; ++m) {
      unsigned em = __shfl(ep, m, 32);
      if (em == 0xFFFFFFFFu) continue;
      unsigned r = em & 0x3FFFu;
      float* ob = out + (size_t)n * kOutPerN + (size_t)r * 136;
      for (int off = lane; off < 136; off += 32) {
        int s = off / 17;
        int v = off - s * 17;
        ob[off] = zst[wave][m * 8 + s] * w3s[v];
      }
    }
  }
}

// ---------------------------------------------------------------------------
extern "C" void kernel_launch(void* const* d_in, const int* in_sizes, int n_in,
                              void* d_out, int out_size, void* d_ws,
                              size_t ws_size, hipStream_t stream) {
  const float* x     = (const float*)d_in[0];
  const int*   val32 = (const int*)d_in[1];   // int64 viewed as 2x i32
  const int*   dep32 = (const int*)d_in[2];   // int64 viewed as 2x i32
  // d_in[3] = pos (unused by reference)
  const float* W1 = (const float*)d_in[4];
  const float* b1 = (const float*)d_in[5];
  const float* W2 = (const float*)d_in[6];
  const float* b2 = (const float*)d_in[7];
  const float* W3 = (const float*)d_in[8];
  float* out = (float*)d_out;

  // workspace layout
  float*    wf     = (float*)d_ws;                               // 1 MiB
  float*    bias8  = (float*)((char*)d_ws + 1048576);            // 32 B
  unsigned* bins   = (unsigned*)((char*)d_ws + 1048640);         // 512 KiB
  unsigned* bincnt = (unsigned*)((char*)d_ws + 1048640 + 524288);// 256 B

  scan_kernel<<<dim3(kN), dim3(256), 0, stream>>>(val32, dep32, bins, bincnt);
  fuse_kernel<<<dim3(1024), dim3(256), 0, stream>>>(W1, b1, W2, b2, wf, bias8);
  {
    long long total = (long long)kN * kOutPerN;
    int blocks = (int)((total + 255) / 256);
    fill_kernel<<<dim3(blocks), dim3(256), 0, stream>>>(b2, W3, out);
  }
  main_kernel<<<dim3(64, 8), dim3(256), 0, stream>>>(x, wf, bias8, W3, bins,
                                                     bincnt, out);
}